// FreqEncoder_8581344657661
// MI455X (gfx1250) — compile-verified
//
#include <hip/hip_runtime.h>
#include <stdint.h>

// FreqEncoder for MI455X (gfx1250):
//  - whole feature table (48 grids x 8 ch x 128 res = 192 KB) staged in LDS,
//    transposed to [n][res][ch] (row = 8 floats = 32 B, 16 B aligned) via
//    CDNA5 async global->LDS b32 loads (per-lane LDS scatter = free transpose),
//    drained with s_wait_asynccnt + workgroup barrier.
//  - compute: one point per thread; 24 v_sin/v_cos; per (scale,phase,dim) the
//    two interp rows are fetched as 2x ds_load_b128 each; outputs written as
//    3-dword (b96-mergeable) contiguous stores per channel.

#define N_SCALES 8
#define X_DIM    3
#define N_CH     8
#define RESN     128
#define N_GRIDS  48                    // N_SCALES * 2 * X_DIM
#define ROW_DW   8                     // floats per (grid,res) row in LDS
#define GRID_DW  (RESN * ROW_DW)       // 1024 dwords per grid
#define TBL_DW   (N_GRIDS * GRID_DW)   // 49152 dwords = 192 KiB
#define OUTW     387                   // 3 + 8*48
#define TPB      1024

#if defined(__AMDGCN__)
typedef __attribute__((address_space(3))) int lds_int_t;
typedef __attribute__((address_space(1))) int glb_int_t;
#endif

__device__ __forceinline__ void async_ld_dword_to_lds(const float* g, uint32_t lds_byte_off) {
#if defined(__AMDGCN__)
#if __has_builtin(__builtin_amdgcn_global_load_async_to_lds_b32)
    __builtin_amdgcn_global_load_async_to_lds_b32(
        (glb_int_t*)(uintptr_t)g, (lds_int_t*)(uintptr_t)lds_byte_off, 0, 0);
#else
    asm volatile("global_load_async_to_lds_b32 %0, %1, off"
                 :: "v"(lds_byte_off), "v"(g)
                 : "memory");
#endif
#endif
}

__device__ __forceinline__ void wait_async_zero() {
#if defined(__AMDGCN__)
#if __has_builtin(__builtin_amdgcn_s_wait_asynccnt)
    __builtin_amdgcn_s_wait_asynccnt(0);
#else
    asm volatile("s_wait_asynccnt 0" ::: "memory");
#endif
#endif
}

__global__ __launch_bounds__(TPB) void freqenc_kernel(
    const float* __restrict__ x,
    const float* __restrict__ feat,
    float* __restrict__ out,
    int npts)
{
    // float4 declaration guarantees 16 B alignment for ds_load_b128 row reads
    __shared__ float4 tbl4[TBL_DW / 4];
    const float* tbl = (const float*)tbl4;

#if defined(__AMDGCN__)
    const uint32_t lds_base = (uint32_t)(uintptr_t)(__attribute__((address_space(3))) float4*)tbl4;
#else
    const uint32_t lds_base = 0;
#endif

    const int tid = threadIdx.x;

    // --- Phase A: async transposed copy of the whole table into LDS ---
    // global element k: n = k/1024, c = (k/128)%8, i = k%128   (layout [n][c][i])
    // LDS target (dwords): n*1024 + i*8 + c                    (layout [n][i][c])
    for (int k = tid; k < N_GRIDS * N_CH * RESN; k += TPB) {
        const int n   = k >> 10;
        const int rem = k & 1023;
        const int c   = rem >> 7;
        const int i   = rem & 127;
        const uint32_t off = lds_base + 4u * (uint32_t)((n << 10) + (i << 3) + c);
        async_ld_dword_to_lds(feat + k, off);
    }
    wait_async_zero();   // each wave drains its own ASYNCcnt
    __syncthreads();     // whole workgroup now sees the LDS table

    // --- Phase B: one point per thread ---
    const int p = (int)(blockIdx.x * TPB + tid);
    if (p >= npts) return;

    const float* xp = x + (size_t)p * 3;
    const float xv[3] = { xp[0], xp[1], xp[2] };
    float* op = out + (size_t)p * OUTW;
    op[0] = xv[0]; op[1] = xv[1]; op[2] = xv[2];

    #pragma unroll
    for (int s = 0; s < N_SCALES; ++s) {
        // scales = 2 ** linspace(0, 8, 8) -> 2^(s*8/7), constant-folded
        const float sc = exp2f((float)s * (8.0f / 7.0f));

        #pragma unroll
        for (int ph = 0; ph < 2; ++ph) {
            float vv[3], fr[3];
            float f0v[3][N_CH], f1v[3][N_CH];

            #pragma unroll
            for (int d = 0; d < X_DIM; ++d) {
                const float t = xv[d] * sc;
                const float v = ph ? __cosf(t) : __sinf(t);
                vv[d] = v;

                // align_corners=True unnormalize: pos = (v+1)*0.5*(res-1)
                const float pos = (v + 1.0f) * 63.5f;
                const float i0f = floorf(pos);
                fr[d] = pos - i0f;
                int i0 = (int)i0f;
                i0 = i0 < 0 ? 0 : (i0 > RESN - 1 ? RESN - 1 : i0);
                int i1 = i0 + 1;
                i1 = i1 > RESN - 1 ? RESN - 1 : i1;

                const int n = s * 6 + ph * 3 + d;
                const float4* r0 = (const float4*)(tbl + (n << 10) + (i0 << 3));
                const float4* r1 = (const float4*)(tbl + (n << 10) + (i1 << 3));
                const float4 a0 = r0[0], a1 = r0[1];   // ds_load_b128 x2
                const float4 b0 = r1[0], b1 = r1[1];   // ds_load_b128 x2
                f0v[d][0] = a0.x; f0v[d][1] = a0.y; f0v[d][2] = a0.z; f0v[d][3] = a0.w;
                f0v[d][4] = a1.x; f0v[d][5] = a1.y; f0v[d][6] = a1.z; f0v[d][7] = a1.w;
                f1v[d][0] = b0.x; f1v[d][1] = b0.y; f1v[d][2] = b0.z; f1v[d][3] = b0.w;
                f1v[d][4] = b1.x; f1v[d][5] = b1.y; f1v[d][6] = b1.z; f1v[d][7] = b1.w;
            }

            // per channel: 3 consecutive dwords -> global_store_b96-mergeable
            #pragma unroll
            for (int c = 0; c < N_CH; ++c) {
                float* dst = op + 3 + c * N_GRIDS + s * 6 + ph * 3;
                #pragma unroll
                for (int d = 0; d < X_DIM; ++d) {
                    const float f0 = f0v[d][c];
                    const float f1 = f1v[d][c];
                    dst[d] = fmaf(f1 - f0, fr[d], f0) + vv[d];
                }
            }
        }
    }
}

extern "C" void kernel_launch(void* const* d_in, const int* in_sizes, int n_in,
                              void* d_out, int out_size, void* d_ws, size_t ws_size,
                              hipStream_t stream) {
    const float* x    = (const float*)d_in[0];   // [P, 3] f32
    const float* feat = (const float*)d_in[1];   // [48, 8, 128] f32
    float* out = (float*)d_out;                  // [P, 387] f32

    const int npts = in_sizes[0] / 3;            // 262144
    const int blocks = (npts + TPB - 1) / TPB;   // 256
    freqenc_kernel<<<blocks, TPB, 0, stream>>>(x, feat, out, npts);
}